// DynamicSpatialConvModule_68238440399455
// MI455X (gfx1250) — compile-verified
//
#include <hip/hip_runtime.h>
#include <hip/hip_bf16.h>
#include <math.h>

typedef float v2f __attribute__((ext_vector_type(2)));
typedef float v8f __attribute__((ext_vector_type(8)));

#define B_  32
#define C_  256
#define H_  56
#define W_  56
#define HW_ 3136           // 56*56
#define K_REMOVE 1568      // round(3136*0.5)

// ---------------------------------------------------------------------------
// Kernel 1: channel pooling (max + mean over 256 channels).
// One wave handles 16 consecutive spatial positions of one batch image.
// Sum via V_WMMA_F32_16X16X4_F32 with B = ones; max via VALU (max3) in the
// same loop; x is streamed once -> non-temporal loads keep L2 clean.
// A-matrix 16x4 f32 layout: lane L -> M = L&15; VGPR0/1 -> K = 2*(L>>4)+{0,1}.
// C/D layout: c[r]: lanes 0-15 -> M=r, lanes 16-31 -> M=8+r (all N columns equal).
// ---------------------------------------------------------------------------
__global__ __launch_bounds__(128)
void pool_kernel(const float* __restrict__ x, float* __restrict__ y2)
{
    const int lane = threadIdx.x & 31;
    const int gwave = blockIdx.x * 4 + (threadIdx.x >> 5);   // 6272 waves total
    const int b    = gwave / (HW_ / 16);                      // 196 tiles per batch
    const int tile = gwave % (HW_ / 16);
    const int hw0  = tile * 16;
    const int pos  = lane & 15;
    const int half = lane >> 4;

    const float* xb = x + (size_t)b * C_ * HW_ + hw0 + pos;

    v8f acc = {};
    v2f bones = {1.0f, 1.0f};
    float mx = -INFINITY;

    for (int c0 = 0; c0 < C_; c0 += 4) {
        const int c = c0 + 2 * half;
        float a0 = __builtin_nontemporal_load(&xb[(size_t)c * HW_]);
        float a1 = __builtin_nontemporal_load(&xb[(size_t)(c + 1) * HW_]);
        v2f amat = {a0, a1};
        acc = __builtin_amdgcn_wmma_f32_16x16x4_f32(
            false, amat, false, bones, (short)0, acc, false, false);
        mx = fmaxf(mx, fmaxf(a0, a1));
    }

    // combine the two channel-halves for the max
    mx = fmaxf(mx, __shfl_xor(mx, 16, 32));

    float* ymax = y2 + (size_t)b * 2 * HW_;   // channel 0 = max
    float* yavg = ymax + HW_;                 // channel 1 = avg

    if (lane < 16)
        ymax[hw0 + lane] = mx;

    // sum extraction: lanes 0..7 -> positions 0..7 (c[lane]); lanes 16..23 -> 8..15 (c[lane&7])
    int j = lane & 7;
    float s = acc[0];
    if (j == 1) s = acc[1];
    if (j == 2) s = acc[2];
    if (j == 3) s = acc[3];
    if (j == 4) s = acc[4];
    if (j == 5) s = acc[5];
    if (j == 6) s = acc[6];
    if (j == 7) s = acc[7];

    bool wAvg = (lane < 8) || (lane >= 16 && lane < 24);
    int apos  = (lane < 8) ? lane : (lane - 8);
    if (wAvg)
        yavg[hw0 + apos] = s * (1.0f / 256.0f);
}

// ---------------------------------------------------------------------------
// Kernel 2: 3x3 conv (2 -> 1, pad 1, stride 1) + sigmoid on pooled maps.
// Pooled maps (800 KB) are L2-resident.
// ---------------------------------------------------------------------------
__global__ __launch_bounds__(256)
void conv_kernel(const float* __restrict__ y2, const float* __restrict__ w,
                 float* __restrict__ yc)
{
    int idx = blockIdx.x * blockDim.x + threadIdx.x;   // b*HW + hw, exact grid
    int b  = idx / HW_;
    int hw = idx % HW_;
    int h  = hw / W_;
    int wq = hw % W_;

    const float* p0 = y2 + (size_t)b * 2 * HW_;   // max plane
    float s = 0.0f;
#pragma unroll
    for (int i = 0; i < 2; ++i) {
        const float* pl = p0 + i * HW_;
#pragma unroll
        for (int kh = 0; kh < 3; ++kh) {
            int hh = h + kh - 1;
            if (hh < 0 || hh >= H_) continue;
#pragma unroll
            for (int kw = 0; kw < 3; ++kw) {
                int ww = wq + kw - 1;
                if (ww < 0 || ww >= W_) continue;
                s += pl[hh * W_ + ww] * w[i * 9 + kh * 3 + kw];
            }
        }
    }
    yc[idx] = 1.0f / (1.0f + expf(-s));
}

// ---------------------------------------------------------------------------
// Kernel 3: per-row exact top-k zeroing via MSB-first radix select.
// One 256-thread block per batch row; row cached in LDS.
// Keys are positive floats -> raw IEEE bits are order-preserving as uint.
// Zero all key < T plus the first (K - count_less) keys == T in index order
// (matches stable argsort tie-breaking).
// ---------------------------------------------------------------------------
__global__ __launch_bounds__(256)
void select_kernel(float* __restrict__ yc)
{
    __shared__ float    vals[HW_];
    __shared__ unsigned hist[256];
    __shared__ unsigned s_sel[2];

    const int tid = threadIdx.x;
    float* row = yc + (size_t)blockIdx.x * HW_;

    for (int i = tid; i < HW_; i += 256) vals[i] = row[i];
    __syncthreads();

    unsigned prefix = 0, maskhi = 0;
    unsigned want = K_REMOVE - 1;     // 0-indexed rank of threshold value T
    unsigned count_less = 0;

    for (int shift = 24; shift >= 0; shift -= 8) {
        hist[tid] = 0;
        __syncthreads();
        for (int i = tid; i < HW_; i += 256) {
            unsigned key = __float_as_uint(vals[i]);
            if ((key & maskhi) == prefix)
                atomicAdd(&hist[(key >> shift) & 255u], 1u);
        }
        __syncthreads();
        if (tid == 0) {
            unsigned run = 0, d = 0;
            for (; d < 256; ++d) {
                unsigned hct = hist[d];
                if (run + hct > want) break;
                run += hct;
            }
            s_sel[0] = d;
            s_sel[1] = run;
        }
        __syncthreads();
        unsigned d = s_sel[0];
        count_less += s_sel[1];
        want       -= s_sel[1];
        prefix |= d << shift;
        maskhi |= 0xFFu << shift;
        __syncthreads();
    }

    const unsigned T = prefix;
    const unsigned quota = K_REMOVE - count_less;   // >= 1

    for (int i = tid; i < HW_; i += 256) {
        if (__float_as_uint(vals[i]) < T) vals[i] = 0.0f;
    }
    __syncthreads();
    if (tid == 0) {
        int q = (int)quota;
        for (int i = 0; i < HW_ && q > 0; ++i) {
            if (__float_as_uint(vals[i]) == T) { vals[i] = 0.0f; --q; }
        }
    }
    __syncthreads();
    for (int i = tid; i < HW_; i += 256) row[i] = vals[i];
}

// ---------------------------------------------------------------------------
// Kernel 4: broadcast masked row to 256 channels.
// Row (12.5 KB) staged in LDS once; stores go LDS -> HBM via CDNA5
// global_store_async_from_lds_b128 (ASYNCcnt), 512B per wave-instruction,
// with TH_STORE_NT since the output is write-once / never re-read.
// Grid: 32 batches x 16 channel-groups; each wave handles 2 channels.
// 3136 floats = 24 full b128 wave-stores + one half-EXEC b128 per channel.
// ---------------------------------------------------------------------------
__global__ __launch_bounds__(256)
void bcast_kernel(const float* __restrict__ yc, float* __restrict__ out)
{
    __shared__ float buf[HW_];

    const int b  = blockIdx.x >> 4;          // /16
    const int cg = blockIdx.x & 15;          // channel group (16 channels)
    const int wave = threadIdx.x >> 5;
    const int lane = threadIdx.x & 31;

    const float* row = yc + (size_t)b * HW_;
    for (int i = threadIdx.x; i < HW_ / 4; i += 256)
        ((float4*)buf)[i] = ((const float4*)row)[i];
    __syncthreads();   // LDS visible to all waves

#pragma unroll
    for (int cc = 0; cc < 2; ++cc) {
        const int c = cg * 16 + wave * 2 + cc;
        float* gbase = out + ((size_t)b * C_ + c) * HW_;
#pragma unroll 4
        for (int jq = 0; jq < 24; ++jq) {
            float*   ga = gbase + jq * 128 + lane * 4;
            unsigned lo = (unsigned)(size_t)(&buf[jq * 128 + lane * 4]);
            asm volatile("global_store_async_from_lds_b128 %0, %1, off th:TH_STORE_NT"
                         :: "v"(ga), "v"(lo) : "memory");
        }
        if (lane < 16) {   // tail: 64 floats, EXEC-masked lanes skip the store
            float*   ga = gbase + 3072 + lane * 4;
            unsigned lo = (unsigned)(size_t)(&buf[3072 + lane * 4]);
            asm volatile("global_store_async_from_lds_b128 %0, %1, off th:TH_STORE_NT"
                         :: "v"(ga), "v"(lo) : "memory");
        }
    }
    asm volatile("s_wait_asynccnt 0" ::: "memory");
}

// ---------------------------------------------------------------------------
extern "C" void kernel_launch(void* const* d_in, const int* in_sizes, int n_in,
                              void* d_out, int out_size, void* d_ws, size_t ws_size,
                              hipStream_t stream)
{
    const float* x = (const float*)d_in[0];   // [32,256,56,56]
    const float* w = (const float*)d_in[1];   // [1,2,3,3]
    float* out = (float*)d_out;               // [32,256,56,56]

    float* y2 = (float*)d_ws;                 // [32][2][3136] pooled (max, avg)
    float* yc = y2 + (size_t)B_ * 2 * HW_;    // [32][3136] conv+sigmoid, then masked

    // 1) pooling: 32*196 = 6272 waves, 4 waves/block
    pool_kernel<<<dim3(B_ * (HW_ / 16) / 4), dim3(128), 0, stream>>>(x, y2);
    // 2) conv 3x3 + sigmoid: 32*3136 threads exactly
    conv_kernel<<<dim3(B_ * HW_ / 256), dim3(256), 0, stream>>>(y2, w, yc);
    // 3) per-row radix-select masking
    select_kernel<<<dim3(B_), dim3(256), 0, stream>>>(yc);
    // 4) broadcast to 256 channels with async NT LDS->global stores
    bcast_kernel<<<dim3(B_ * 16), dim3(256), 0, stream>>>(yc, out);
}